// MultiHeadAttention_15367392985397
// MI455X (gfx1250) — compile-verified
//
#include <hip/hip_runtime.h>
#include <hip/hip_bf16.h>

// Problem constants (fixed by reference)
#define BB 2
#define SS 2048
#define DD 1024
#define NH 16
#define DH 64

typedef __attribute__((ext_vector_type(2))) float v2f;
typedef __attribute__((ext_vector_type(8))) float v8f;

__device__ __forceinline__ v8f v8f_zero() {
    v8f z = {0.f, 0.f, 0.f, 0.f, 0.f, 0.f, 0.f, 0.f};
    return z;
}

__device__ __forceinline__ v2f ld2(const float* p) {
    return *reinterpret_cast<const v2f*>(p);
}

// D = A(16x4 f32) x B(4x16 f32) + C  -- native fp32 WMMA
__device__ __forceinline__ v8f wmma4(v2f a, v2f b, v8f c) {
    return __builtin_amdgcn_wmma_f32_16x16x4_f32(
        /*neg_a=*/false, a, /*neg_b=*/false, b,
        /*c_mod=*/(short)0, c, /*reuse_a=*/false, /*reuse_b=*/false);
}

// ---------------------------------------------------------------------------
// Generic NT GEMM: Y = X (M x K) * W^T, W is (N x K) row-major (nn.Linear).
// Per wave: 32(M) x 64(N) tile, 2x4 accumulator blocking.
// HEADSPLIT=1: Y written as (B*NH, S, DH) head-unpacked.  M = B*S, N = D.
// HEADSPLIT=0: Y written row-major (M x N).
// ---------------------------------------------------------------------------
template <int HEADSPLIT>
__global__ void gemm_nt_kernel(const float* __restrict__ X,
                               const float* __restrict__ W,
                               float* __restrict__ Y) {
    const int lane  = threadIdx.x & 31;
    const int gwid  = (blockIdx.x * blockDim.x + threadIdx.x) >> 5;
    const int MT    = (BB * SS) / 32;          // 128 M-tiles
    const int m_base = (gwid % MT) * 32;
    const int n_base = (gwid / MT) * 64;
    const int frow  = lane & 15;               // fragment row (M or N)
    const int ksel  = (lane >> 4) << 1;        // 0 or 2

    v8f acc[2][4];
#pragma unroll
    for (int i = 0; i < 2; ++i)
#pragma unroll
        for (int j = 0; j < 4; ++j) acc[i][j] = v8f_zero();

    const float* x0 = X + (size_t)(m_base + frow) * DD;
    const float* x1 = x0 + (size_t)16 * DD;
    const float* w0 = W + (size_t)(n_base + frow) * DD;
    const float* w1 = w0 + (size_t)16 * DD;
    const float* w2 = w0 + (size_t)32 * DD;
    const float* w3 = w0 + (size_t)48 * DD;

#pragma unroll 4
    for (int k = 0; k < DD; k += 4) {
        const int ko = k + ksel;
        v2f a0 = ld2(x0 + ko);
        v2f a1 = ld2(x1 + ko);
        v2f b0 = ld2(w0 + ko);
        v2f b1 = ld2(w1 + ko);
        v2f b2 = ld2(w2 + ko);
        v2f b3 = ld2(w3 + ko);
        acc[0][0] = wmma4(a0, b0, acc[0][0]);
        acc[0][1] = wmma4(a0, b1, acc[0][1]);
        acc[0][2] = wmma4(a0, b2, acc[0][2]);
        acc[0][3] = wmma4(a0, b3, acc[0][3]);
        acc[1][0] = wmma4(a1, b0, acc[1][0]);
        acc[1][1] = wmma4(a1, b1, acc[1][1]);
        acc[1][2] = wmma4(a1, b2, acc[1][2]);
        acc[1][3] = wmma4(a1, b3, acc[1][3]);
    }

    const int rofs = (lane >> 4) << 3;         // rows 0..7 or 8..15
    const int ncol = lane & 15;
#pragma unroll
    for (int i = 0; i < 2; ++i) {
#pragma unroll
        for (int j = 0; j < 4; ++j) {
#pragma unroll
            for (int r = 0; r < 8; ++r) {
                const int m = m_base + i * 16 + r + rofs;
                const int n = n_base + j * 16 + ncol;
                const float val = acc[i][j][r];
                if (HEADSPLIT) {
                    const int b = m >> 11, s = m & (SS - 1);
                    const int h = n >> 6, d = n & (DH - 1);
                    Y[(((size_t)(b * NH + h) * SS) + s) * DH + d] = val;
                } else {
                    Y[(size_t)m * DD + n] = val;
                }
            }
        }
    }
}

// ---------------------------------------------------------------------------
// scores = Qh Kh^T / 8, masked (key >= valid_len -> -1e6). Per wave 32x64.
// Qh,Kh: (B*NH, S, DH).  attn: (B*NH, S, S) written into d_out region.
// ---------------------------------------------------------------------------
__global__ void scores_kernel(const float* __restrict__ Qh,
                              const float* __restrict__ Kh,
                              const int* __restrict__ valid_len,
                              float* __restrict__ attn) {
    const int lane = threadIdx.x & 31;
    const int gwid = (blockIdx.x * blockDim.x + threadIdx.x) >> 5;
    const int head = gwid / 2048;              // 32 heads, 2048 waves each
    const int rem  = gwid % 2048;
    const int m_base = (rem & 63) * 32;        // 64 M-tiles of 32
    const int n_base = (rem >> 6) * 64;        // 32 N-tiles of 64
    const int frow = lane & 15;
    const int ksel = (lane >> 4) << 1;

    v8f acc[2][4];
#pragma unroll
    for (int i = 0; i < 2; ++i)
#pragma unroll
        for (int j = 0; j < 4; ++j) acc[i][j] = v8f_zero();

    const float* q0 = Qh + ((size_t)head * SS + m_base + frow) * DH;
    const float* q1 = q0 + (size_t)16 * DH;
    const float* k0 = Kh + ((size_t)head * SS + n_base + frow) * DH;
    const float* k1 = k0 + (size_t)16 * DH;
    const float* k2 = k0 + (size_t)32 * DH;
    const float* k3 = k0 + (size_t)48 * DH;

#pragma unroll
    for (int k = 0; k < DH; k += 4) {
        const int ko = k + ksel;
        v2f a0 = ld2(q0 + ko);
        v2f a1 = ld2(q1 + ko);
        v2f b0 = ld2(k0 + ko);
        v2f b1 = ld2(k1 + ko);
        v2f b2 = ld2(k2 + ko);
        v2f b3 = ld2(k3 + ko);
        acc[0][0] = wmma4(a0, b0, acc[0][0]);
        acc[0][1] = wmma4(a0, b1, acc[0][1]);
        acc[0][2] = wmma4(a0, b2, acc[0][2]);
        acc[0][3] = wmma4(a0, b3, acc[0][3]);
        acc[1][0] = wmma4(a1, b0, acc[1][0]);
        acc[1][1] = wmma4(a1, b1, acc[1][1]);
        acc[1][2] = wmma4(a1, b2, acc[1][2]);
        acc[1][3] = wmma4(a1, b3, acc[1][3]);
    }

    const int vl = valid_len[head >> 4];       // repeat_interleave by NH
    const int rofs = (lane >> 4) << 3;
    const int ncol = lane & 15;
    float* ah = attn + (size_t)head * SS * SS;
#pragma unroll
    for (int i = 0; i < 2; ++i) {
#pragma unroll
        for (int j = 0; j < 4; ++j) {
#pragma unroll
            for (int r = 0; r < 8; ++r) {
                const int m = m_base + i * 16 + r + rofs;
                const int n = n_base + j * 16 + ncol;
                float val = acc[i][j][r] * 0.125f;  // 1/sqrt(64)
                if (n >= vl) val = -1e6f;
                ah[(size_t)m * SS + n] = val;
            }
        }
    }
}

// ---------------------------------------------------------------------------
// In-place row softmax over attn rows of length S. One 256-thread block/row.
// ---------------------------------------------------------------------------
__global__ void softmax_kernel(float* __restrict__ attn) {
    const size_t row = blockIdx.x;
    float* p = attn + row * (size_t)SS;
    const int t = threadIdx.x;
    float4* p4 = reinterpret_cast<float4*>(p) + (t * 2);
    float4 x0 = p4[0];
    float4 x1 = p4[1];

    float m = fmaxf(fmaxf(fmaxf(x0.x, x0.y), fmaxf(x0.z, x0.w)),
                    fmaxf(fmaxf(x1.x, x1.y), fmaxf(x1.z, x1.w)));
#pragma unroll
    for (int off = 16; off > 0; off >>= 1) m = fmaxf(m, __shfl_xor(m, off, 32));
    __shared__ float smax[8];
    if ((t & 31) == 0) smax[t >> 5] = m;
    __syncthreads();
    m = smax[0];
#pragma unroll
    for (int i = 1; i < 8; ++i) m = fmaxf(m, smax[i]);

    x0.x = __expf(x0.x - m); x0.y = __expf(x0.y - m);
    x0.z = __expf(x0.z - m); x0.w = __expf(x0.w - m);
    x1.x = __expf(x1.x - m); x1.y = __expf(x1.y - m);
    x1.z = __expf(x1.z - m); x1.w = __expf(x1.w - m);

    float s = (x0.x + x0.y + x0.z + x0.w) + (x1.x + x1.y + x1.z + x1.w);
#pragma unroll
    for (int off = 16; off > 0; off >>= 1) s += __shfl_xor(s, off, 32);
    __shared__ float ssum[8];
    if ((t & 31) == 0) ssum[t >> 5] = s;
    __syncthreads();
    float tot = 0.f;
#pragma unroll
    for (int i = 0; i < 8; ++i) tot += ssum[i];
    const float inv = 1.0f / tot;

    x0.x *= inv; x0.y *= inv; x0.z *= inv; x0.w *= inv;
    x1.x *= inv; x1.y *= inv; x1.z *= inv; x1.w *= inv;
    p4[0] = x0;
    p4[1] = x1;
}

// ---------------------------------------------------------------------------
// outh = attn @ Vh, written in merged layout merged[b, s, d*NH + h].
// Per wave: 32(M=seq) x 64(N=DH full). 64 waves/head.
// ---------------------------------------------------------------------------
__global__ void av_kernel(const float* __restrict__ attn,
                          const float* __restrict__ Vh,
                          float* __restrict__ merged) {
    const int lane = threadIdx.x & 31;
    const int gwid = (blockIdx.x * blockDim.x + threadIdx.x) >> 5;
    const int head = gwid >> 6;
    const int m_base = (gwid & 63) * 32;
    const int b = head >> 4, h = head & 15;
    const int frow = lane & 15;
    const int ksel = (lane >> 4) << 1;
    const int ncol = lane & 15;

    v8f acc[2][4];
#pragma unroll
    for (int i = 0; i < 2; ++i)
#pragma unroll
        for (int j = 0; j < 4; ++j) acc[i][j] = v8f_zero();

    const float* a0p = attn + (size_t)head * SS * SS + (size_t)(m_base + frow) * SS;
    const float* a1p = a0p + (size_t)16 * SS;
    const float* vb  = Vh + (size_t)head * SS * DH;

#pragma unroll 2
    for (int k = 0; k < SS; k += 4) {
        const int ko = k + ksel;
        v2f a0 = ld2(a0p + ko);
        v2f a1 = ld2(a1p + ko);
        const float* vp = vb + (size_t)ko * DH;
        v2f b0, b1, b2, b3;
        b0.x = vp[ncol];      b0.y = vp[DH + ncol];
        b1.x = vp[16 + ncol]; b1.y = vp[DH + 16 + ncol];
        b2.x = vp[32 + ncol]; b2.y = vp[DH + 32 + ncol];
        b3.x = vp[48 + ncol]; b3.y = vp[DH + 48 + ncol];
        acc[0][0] = wmma4(a0, b0, acc[0][0]);
        acc[0][1] = wmma4(a0, b1, acc[0][1]);
        acc[0][2] = wmma4(a0, b2, acc[0][2]);
        acc[0][3] = wmma4(a0, b3, acc[0][3]);
        acc[1][0] = wmma4(a1, b0, acc[1][0]);
        acc[1][1] = wmma4(a1, b1, acc[1][1]);
        acc[1][2] = wmma4(a1, b2, acc[1][2]);
        acc[1][3] = wmma4(a1, b3, acc[1][3]);
    }

    const int rofs = (lane >> 4) << 3;
#pragma unroll
    for (int i = 0; i < 2; ++i) {
#pragma unroll
        for (int j = 0; j < 4; ++j) {
#pragma unroll
            for (int r = 0; r < 8; ++r) {
                const int s = m_base + i * 16 + r + rofs;     // seq index
                const int d = j * 16 + ncol;                  // head dim
                // merge_base_head: merged[b, s, d*NH + h]
                merged[((size_t)b * SS + s) * DD + d * NH + h] = acc[i][j][r];
            }
        }
    }
}

// ---------------------------------------------------------------------------
extern "C" void kernel_launch(void* const* d_in, const int* in_sizes, int n_in,
                              void* d_out, int out_size, void* d_ws, size_t ws_size,
                              hipStream_t stream) {
    const float* q  = (const float*)d_in[0];
    const float* k  = (const float*)d_in[1];
    const float* v  = (const float*)d_in[2];
    const float* Wq = (const float*)d_in[3];
    const float* Wk = (const float*)d_in[4];
    const float* Wv = (const float*)d_in[5];
    const float* Wo = (const float*)d_in[6];
    const int*   vl = (const int*)d_in[7];

    float* out  = (float*)d_out;                          // (B, S, D)
    float* attn = out + (size_t)BB * SS * DD;             // (B*NH, S, S)

    const size_t HSZ = (size_t)BB * NH * SS * DH;         // 4,194,304 floats
    float* ws = (float*)d_ws;
    float* qh = ws;
    float* kh = ws + HSZ;
    float* vh = ws + 2 * HSZ;
    float* merged = ws;                                   // reuse qh after scores

    dim3 blk(256);

    // Projections, fused head-unpack: 2048 waves each -> 256 blocks of 8 waves.
    gemm_nt_kernel<1><<<256, blk, 0, stream>>>(q, Wq, qh);
    gemm_nt_kernel<1><<<256, blk, 0, stream>>>(k, Wk, kh);
    gemm_nt_kernel<1><<<256, blk, 0, stream>>>(v, Wv, vh);

    // Scores + mask: 32 heads * 2048 waves = 65536 waves -> 8192 blocks.
    scores_kernel<<<8192, blk, 0, stream>>>(qh, kh, vl, attn);

    // Row softmax in place: one block per row (32 * 2048 rows).
    softmax_kernel<<<BB * NH * SS, blk, 0, stream>>>(attn);

    // attn @ V with fused merge: 2048 waves -> 256 blocks.
    av_kernel<<<256, blk, 0, stream>>>(attn, vh, merged);

    // Output projection: 2048 waves -> 256 blocks.
    gemm_nt_kernel<0><<<256, blk, 0, stream>>>(merged, Wo, out);
}